// YOLOLoss_42073499632029
// MI455X (gfx1250) — compile-verified
//
#include <hip/hip_runtime.h>

#define NCLS 80
#define EPSF 1e-6f
#define NBLK 256   // partial-sum blocks per scale (multiple of 64 for WMMA reduce)

typedef float v2f __attribute__((ext_vector_type(2)));
typedef float v8f __attribute__((ext_vector_type(8)));

__constant__ float c_anch[3][3][2] = {
  {{10.f/640.f,13.f/640.f},{16.f/640.f,30.f/640.f},{33.f/640.f,23.f/640.f}},
  {{30.f/640.f,61.f/640.f},{62.f/640.f,45.f/640.f},{59.f/640.f,119.f/640.f}},
  {{116.f/640.f,90.f/640.f},{156.f/640.f,198.f/640.f},{373.f/640.f,326.f/640.f}},
};

__device__ __forceinline__ float softplusf(float x){
  return fmaxf(x, 0.f) + log1pf(expf(-fabsf(x)));
}
__device__ __forceinline__ float sigmoidf(float x){ return 1.f/(1.f+expf(-x)); }
__device__ __forceinline__ float smoothl1(float x, float t){
  float d = fabsf(x - t);
  return d < 0.1f ? 5.f*d*d : d - 0.05f;   // beta = 0.1
}

// ---- zero the (tiny) workspace accumulators + partial array ----
__global__ void k_zero(float* ws){
  if (threadIdx.x < 3*NBLK + 24) ws[threadIdx.x] = 0.f;
}

// ---- dense: sum softplus(p_obj) over all (b, anchor, h, w) per scale ----
__global__ void k_obj_bulk(const float* __restrict__ p3, const float* __restrict__ p4,
                           const float* __restrict__ p5, float* __restrict__ part){
  int s   = blockIdx.x >> 8;     // NBLK = 256 blocks per scale
  int blk = blockIdx.x & (NBLK-1);
  const float* p = (s==0) ? p3 : ((s==1) ? p4 : p5);
  int HW    = (s==0) ? 6400 : ((s==1) ? 1600 : 400);
  int total = 48 * HW;           // B(16) * anchors(3) * HW
  float acc = 0.f;
  for (int t = blk*256 + threadIdx.x; t < total; t += NBLK*256){
    int ba = t / HW;
    int hw = t - ba*HW;
    int b  = ba / 3;
    int a  = ba - b*3;
    float x = p[((size_t)(b*255 + a*85 + 4))*HW + hw];  // channel k=4 (objectness)
    acc += softplusf(x);
  }
  // wave32 reduce, then cross-wave via LDS
  for (int o = 16; o > 0; o >>= 1) acc += __shfl_xor(acc, o, 32);
  __shared__ float lsum[8];
  int lane = threadIdx.x & 31, wv = threadIdx.x >> 5;
  if (lane == 0) lsum[wv] = acc;
  __syncthreads();
  if (threadIdx.x == 0){
    float t2 = 0.f;
    for (int i = 0; i < 8; i++) t2 += lsum[i];
    part[s*NBLK + blk] = t2;
  }
}

// ---- sparse: target build + masked box/cls/obj-correction terms ----
// one wave per (scale, batch); thread n owns box n. JAX scatter-set dedup:
//   is_first  -> npos, obj correction, class softplus sum (once per distinct cell)
//   is_last   -> t_xy / t_wh winner (last write wins)
//   cls_first -> t_cls union of (cell, class)
__global__ void k_masked(const float* __restrict__ p3, const float* __restrict__ p4,
                         const float* __restrict__ p5, const float* __restrict__ boxes,
                         const int* __restrict__ cls_ids, float* __restrict__ acc){
  int s = blockIdx.x / 16;
  int b = blockIdx.x % 16;
  const float* p = (s==0) ? p3 : ((s==1) ? p4 : p5);
  int H  = (s==0) ? 80 : ((s==1) ? 40 : 20);
  int W  = H, HW = H*W;
  int n  = threadIdx.x;            // 0..31

  __shared__ int skey[32];
  __shared__ int scls[32];

  const float* bx = boxes + ((size_t)b*32 + n)*4;
  float cx = bx[0], cy = bx[1], w = bx[2], h = bx[3];

  float best_s = -1.f; int best_a = 0;
  for (int a = 0; a < 3; a++){
    float aw = c_anch[s][a][0], ah = c_anch[s][a][1];
    float ss  = 1.f/(1.f + fabsf(w-aw) + fabsf(h-ah));
    float asp = 1.f/(1.f + fabsf(w/(h+EPSF) - aw/(ah+EPSF)));
    float sc  = ss*asp;
    if (sc > best_s){ best_s = sc; best_a = a; }   // first-max wins (argmax)
  }
  float area = w*h;
  bool scale_ok = (s==0) ? (area < 0.05f)
                : (s==1) ? (area >= 0.02f && area < 0.15f)
                         : (area >= 0.08f);
  bool assign = (best_s > 0.2f) || scale_ok;       // covers >0.4 case too
  int gx = min(max((int)(cx*W), 0), W-1);
  int gy = min(max((int)(cy*H), 0), H-1);
  int key = assign ? ((best_a*H + gy)*W + gx) : -1;
  int cid = cls_ids[b*32 + n];
  skey[n] = key; scls[n] = cid;
  __syncthreads();

  if (assign){
    bool is_first = true, is_last = true, cls_first = true;
    for (int j = 0; j < 32; j++){
      if (j < n && skey[j] == key){
        is_first = false;
        if (scls[j] == cid) cls_first = false;
      }
      if (j > n && skey[j] == key) is_last = false;
    }
    size_t base = ((size_t)(b*255 + best_a*85))*HW + (size_t)gy*W + gx;
    float* A = acc + s*8;
    if (is_first){
      atomicAdd(&A[0], 1.f);                           // npos
      atomicAdd(&A[1], p[base + 4*(size_t)HW]);        // obj correction (t_obj=1)
      float sp = 0.f;
      for (int k = 0; k < NCLS; k++)
        sp += softplusf(p[base + (size_t)(5+k)*HW]);   // per-cell class softplus
      atomicAdd(&A[4], sp);
    }
    if (cls_first){
      atomicAdd(&A[5], p[base + (size_t)(5+cid)*HW]);  // -x*t term of class BCE
    }
    if (is_last){
      float tx  = cx*W - gx, ty = cy*H - gy;
      float aw  = c_anch[s][best_a][0], ah = c_anch[s][best_a][1];
      float twx = logf(fmaxf(EPSF, w*640.f/(aw*640.f + EPSF)));
      float twy = logf(fmaxf(EPSF, h*640.f/(ah*640.f + EPSF)));
      float px  = sigmoidf(p[base]);
      float py  = sigmoidf(p[base + (size_t)HW]);
      atomicAdd(&A[2], smoothl1(px, tx) + smoothl1(py, ty));
      float pw = p[base + 2*(size_t)HW], ph = p[base + 3*(size_t)HW];
      atomicAdd(&A[3], smoothl1(pw, twx) + smoothl1(ph, twy));
    }
  }
}

// ---- finalize: WMMA ones-matrix reduction of per-block partials + scalar combine ----
__global__ void k_final(const float* __restrict__ part, const float* __restrict__ acc,
                        float* __restrict__ out){
  int lane = threadIdx.x;          // exactly 32 threads, EXEC all ones for WMMA
  float bulk0, bulk1, bulk2;
  float* bulks[3] = {&bulk0, &bulk1, &bulk2};
  v2f ones; ones.x = 1.f; ones.y = 1.f;
  for (int s = 0; s < 3; s++){
    v8f c = {};
    for (int it = 0; it < NBLK/64; it++){
      v2f b2;
      b2.x = part[s*NBLK + it*64 + lane];
      b2.y = part[s*NBLK + it*64 + 32 + lane];
      // D[m,n] = C[m,n] + colsum_B[n]  (A = all ones 16x4)
      c = __builtin_amdgcn_wmma_f32_16x16x4_f32(
            /*neg_a=*/false, ones, /*neg_b=*/false, b2,
            /*c_mod=*/(short)0, c, /*reuse_a=*/false, /*reuse_b=*/false);
    }
    // each lane's c[0] holds one column sum; each column appears exactly twice
    float v = c[0];
    for (int o = 16; o > 0; o >>= 1) v += __shfl_xor(v, o, 32);
    *bulks[s] = 0.5f * v;
  }
  if (lane == 0){
    const float cnt[3] = {307200.f, 76800.f, 19200.f};  // B*3*H*W per scale
    float bulk[3] = {bulk0, bulk1, bulk2};
    float box = 0.f, obj = 0.f, cls = 0.f;
    for (int s = 0; s < 3; s++){
      const float* A = acc + s*8;
      float npos = A[0];
      obj += (bulk[s] - A[1]) / cnt[s];
      if (npos > 0.f){
        box += (A[2] + A[3]) / (2.f*npos);
        cls += (A[4] - A[5]) / (npos * (float)NCLS);
      }
    }
    out[0] = 0.15f*box + obj + 0.6f*cls;
    out[1] = box;
    out[2] = obj;
    out[3] = cls;
  }
}

extern "C" void kernel_launch(void* const* d_in, const int* in_sizes, int n_in,
                              void* d_out, int out_size, void* d_ws, size_t ws_size,
                              hipStream_t stream){
  (void)in_sizes; (void)n_in; (void)out_size; (void)ws_size;
  const float* p3    = (const float*)d_in[0];
  const float* p4    = (const float*)d_in[1];
  const float* p5    = (const float*)d_in[2];
  const float* boxes = (const float*)d_in[3];
  const int*   cls   = (const int*)d_in[4];
  float* ws   = (float*)d_ws;
  float* part = ws;            // 3*NBLK floats
  float* acc  = ws + 3*NBLK;   // 3*8 floats: npos, objcorr, sl1xy, sl1wh, clssp, clscorr
  k_zero    <<<1, 1024, 0, stream>>>(ws);
  k_obj_bulk<<<3*NBLK, 256, 0, stream>>>(p3, p4, p5, part);
  k_masked  <<<48, 32, 0, stream>>>(p3, p4, p5, boxes, cls, acc);
  k_final   <<<1, 32, 0, stream>>>(part, acc, (float*)d_out);
}